// CenterLoss_41240275976790
// MI455X (gfx1250) — compile-verified
//
#include <hip/hip_runtime.h>
#include <math.h>

typedef __attribute__((ext_vector_type(2))) float v2f;
typedef __attribute__((ext_vector_type(8))) float v8f;

#define BATCH 4096
#define FEAT 512
#define ROWS_PER_BLOCK 16
#define WAVES_PER_BLOCK 8
#define K_PER_WAVE (FEAT / WAVES_PER_BLOCK) /* 64 */

// Kernel 1: each block handles 16 batch rows; 8 waves split the 512-wide
// feature dim. Each wave accumulates per-row sum-of-squares via
// V_WMMA_F32_16X16X4_F32 with A = (diff*diff) tile and B = ones, so every
// column of D holds the 16 row sums (full f32 precision, tensor pipe does
// the reduction at zero marginal cost in this bandwidth-bound kernel).
__global__ __launch_bounds__(256) void center_loss_partial(
    const float* __restrict__ x,
    const int*   __restrict__ labels,
    const float* __restrict__ centers,
    float*       __restrict__ block_sums)
{
    __shared__ float partial[WAVES_PER_BLOCK * ROWS_PER_BLOCK];

    const int tid  = threadIdx.x;
    const int wave = tid >> 5;
    const int lane = tid & 31;
    const int m    = lane & 15;          // matrix row this lane feeds (A layout)
    const int kOff = (lane >> 4) << 1;   // lanes 0-15 -> K=0,1 ; lanes 16-31 -> K=2,3
    const int row  = blockIdx.x * ROWS_PER_BLOCK + m;

    const int lab = labels[row];
    const float* __restrict__ xp = x       + (size_t)row * FEAT + kOff;
    const float* __restrict__ cp = centers + (size_t)lab * FEAT + kOff;

    const int kBase = wave * K_PER_WAVE;

    v8f acc = {};
    v2f ones;
    ones[0] = 1.0f;
    ones[1] = 1.0f;

#pragma unroll 4
    for (int k = kBase; k < kBase + K_PER_WAVE; k += 4) {
        v2f xa = *(const v2f*)(xp + k);   // global_load_b64, 8B aligned
        v2f ca = *(const v2f*)(cp + k);   // gathered center row, 8B aligned
        v2f d;
        d[0] = xa[0] - ca[0];
        d[1] = xa[1] - ca[1];
        v2f a;
        a[0] = d[0] * d[0];
        a[1] = d[1] * d[1];
        // D = A(16x4) * ones(4x16) + C  ->  every column = row sums
        acc = __builtin_amdgcn_wmma_f32_16x16x4_f32(
            /*neg_a=*/false, a, /*neg_b=*/false, ones,
            /*c_mod=*/(short)0, acc, /*reuse_a=*/false, /*reuse_b=*/false);
    }

    // C/D layout: VGPR v, lanes 0-15 hold D[v][n] (row v), lanes 16-31 hold row v+8.
    // All columns identical, so lane 0 / lane 16 each carry 8 row partials.
    if (lane == 0) {
#pragma unroll
        for (int i = 0; i < 8; ++i) partial[wave * 16 + i] = acc[i];
    }
    if (lane == 16) {
#pragma unroll
        for (int i = 0; i < 8; ++i) partial[wave * 16 + 8 + i] = acc[i];
    }
    __syncthreads();

    // Deterministic fixed-order combine: sum wave partials per row, sqrt, sum rows.
    if (tid == 0) {
        float t = 0.0f;
#pragma unroll
        for (int r = 0; r < ROWS_PER_BLOCK; ++r) {
            float s = 0.0f;
#pragma unroll
            for (int w = 0; w < WAVES_PER_BLOCK; ++w) s += partial[w * 16 + r];
            t += sqrtf(s);
        }
        block_sums[blockIdx.x] = t;
    }
}

// Kernel 2: fixed-order reduction of the 256 per-block sums -> mean.
__global__ void center_loss_final(const float* __restrict__ block_sums,
                                  float* __restrict__ out, int nblocks)
{
    if (blockIdx.x == 0 && threadIdx.x == 0) {
        float t = 0.0f;
        for (int i = 0; i < nblocks; ++i) t += block_sums[i];
        out[0] = t / (float)BATCH;
    }
}

extern "C" void kernel_launch(void* const* d_in, const int* in_sizes, int n_in,
                              void* d_out, int out_size, void* d_ws, size_t ws_size,
                              hipStream_t stream)
{
    (void)in_sizes; (void)n_in; (void)out_size; (void)ws_size;
    const float* x       = (const float*)d_in[0];
    const int*   labels  = (const int*)d_in[1];
    const float* centers = (const float*)d_in[2];
    float*       out     = (float*)d_out;
    float*       bsums   = (float*)d_ws;   // 256 floats of scratch

    const int nblocks = BATCH / ROWS_PER_BLOCK;  // 256
    center_loss_partial<<<nblocks, 256, 0, stream>>>(x, labels, centers, bsums);
    center_loss_final<<<1, 32, 0, stream>>>(bsums, out, nblocks);
}